// GNNClassification_8022998909047
// MI455X (gfx1250) — compile-verified
//
#include <hip/hip_runtime.h>
#include <math.h>

#define N_NODES   100000
#define N_EDGES   3200000
#define CH        128
#define N_CLASSES 16
#define M_TILES   (N_NODES / 16)   // 6250, exact

typedef float v2f __attribute__((ext_vector_type(2)));
typedef float v8f __attribute__((ext_vector_type(8)));
typedef unsigned int u32x4 __attribute__((ext_vector_type(4)));
typedef int          i32x4 __attribute__((ext_vector_type(4)));
typedef int          i32x8 __attribute__((ext_vector_type(8)));

// ---------------- degree / symmetric norm ----------------
__global__ __launch_bounds__(256) void k_deg_init(float* __restrict__ deg) {
    int i = blockIdx.x * 256 + threadIdx.x;
    if (i < N_NODES) deg[i] = 1.0f;          // self-loop contributes 1
}

__global__ __launch_bounds__(256) void k_deg_scatter(const int* __restrict__ dst,
                                                     float* __restrict__ deg) {
    int e = blockIdx.x * 256 + threadIdx.x;
    if (e < N_EDGES) atomicAdd(&deg[dst[e]], 1.0f);
}

__global__ __launch_bounds__(256) void k_deg_fin(float* __restrict__ deg) {
    int i = blockIdx.x * 256 + threadIdx.x;
    if (i < N_NODES) deg[i] = rsqrtf(deg[i]);   // deg >= 1 always
}

// ---------------- TDM helper: 2-D tile, global -> LDS, optional LDS pad ----------------
// D# layout per CDNA5 ISA §8.3-8.5. Group0: count=1, lds_addr, global_addr, type=2.
// Group1: data_size=4B, pad cfg, tensor/tile dims (tensor dims == tile dims: no OOB),
// row stride in elements. Groups 2/3 zero (<=2D tensor, tile_dim2/3 unused).
// 6-arg builtin form (clang-23 / therock-10.0 lane).
__device__ __forceinline__ void tdm_load_2d(const void* gsrc, unsigned int lds_off,
                                            int d0, int d1, int row_stride,
                                            int pad_interval_code, int pad_amount_code)
{
    unsigned long long ga = (unsigned long long)(uintptr_t)gsrc;
    u32x4 g0;
    g0.x = 1u;                                               // count=1, user descriptor
    g0.y = lds_off;                                          // lds_addr (bytes)
    g0.z = (unsigned int)ga;                                 // global_addr[31:0]
    g0.w = ((unsigned int)(ga >> 32) & 0x01FFFFFFu) | (2u << 30); // addr[56:32] | type=2
    i32x8 g1;
    g1[0] = (2 << 16) | (1 << 20) |                          // data_size=4B, pad_enable
            (pad_interval_code << 22) | (pad_amount_code << 25);
    g1[1] = (d0 & 0xFFFF) << 16;                             // tensor_dim0[15:0]
    g1[2] = ((unsigned)d0 >> 16) | ((d1 & 0xFFFF) << 16);    // tensor_dim0[31:16]|tensor_dim1[15:0]
    g1[3] = ((unsigned)d1 >> 16) | ((unsigned)d0 << 16);     // tensor_dim1[31:16]|tile_dim0
    g1[4] = d1 & 0xFFFF;                                     // tile_dim1 (tile_dim2=0)
    g1[5] = row_stride;                                      // tensor_dim0_stride[31:0]
    g1[6] = 0;
    g1[7] = 0;
    i32x4 z4 = {0, 0, 0, 0};
    i32x8 z8 = {0, 0, 0, 0, 0, 0, 0, 0};
    __builtin_amdgcn_tensor_load_to_lds(g0, g1, z4, z4, z8, 0);
}

__device__ __forceinline__ unsigned int lds_off_of(const void* p) {
    return (unsigned int)(uintptr_t)p;      // addrspace(3) flat addr: low 32 = LDS offset
}

// ---------------- WMMA GEMM: H[n,128] = X[n,128] @ W[128,128] ----------------
#define XS_LD 132   // 128 dwords + 4 pad (TDM pad_interval=6, pad_amount=3)
#define WS_LD 72    // 64 dwords + 8 pad  (TDM pad_interval=5, pad_amount=7)
#define GTPB  5     // M tiles per block

template <int RELU>
__global__ __launch_bounds__(128)
void k_gemm128(const float* __restrict__ X, const float* __restrict__ W,
               float* __restrict__ H)
{
    __shared__ float Ws[128 * WS_LD];       // 36 KB: 64-column group of W
    __shared__ float Xs[2][16 * XS_LD];     // 2 x 8.25 KB: double-buffered X tiles

    const int tid  = threadIdx.x;
    const int wave = tid >> 5;
    const int lane = tid & 31;
    const int half = lane >> 4;             // K-split group per WMMA layout
    const int m    = lane & 15;
    const int cg   = blockIdx.y;            // column group: cols [cg*64, cg*64+64)

    const int n_in_grp = wave * 16 + m;     // 0..63 within column group
    const int t0 = blockIdx.x * GTPB;
    const int t1 = (t0 + GTPB < M_TILES) ? (t0 + GTPB) : M_TILES;

    if (wave == 0) {
        // W column group: 64 wide x 128 rows, row stride CH, pad 64dw->+8dw (WS_LD=72)
        tdm_load_2d(W + cg * 64, lds_off_of(Ws), 64, 128, CH, 5, 7);
        // first X tile: 128 wide x 16 rows, pad 128dw->+4dw (XS_LD=132)
        tdm_load_2d(X + (size_t)t0 * 16 * CH, lds_off_of(&Xs[0][0]), 128, 16, CH, 6, 3);
    }

    for (int t = t0; t < t1; ++t) {
        const int buf = (t - t0) & 1;
        if (wave == 0) __builtin_amdgcn_s_wait_tensorcnt(0);  // current buf (+Ws) landed
        __syncthreads();                                      // publish to all waves
        if (wave == 0 && (t + 1) < t1)                        // prefetch next tile (overlaps)
            tdm_load_2d(X + (size_t)(t + 1) * 16 * CH,
                        lds_off_of(&Xs[buf ^ 1][0]), 128, 16, CH, 6, 3);

        v8f acc = {0.f, 0.f, 0.f, 0.f, 0.f, 0.f, 0.f, 0.f};
        #pragma unroll
        for (int k0 = 0; k0 < 128; k0 += 4) {
            const int ka = k0 + half * 2;   // lanes 0-15: K=k0,k0+1; 16-31: k0+2,k0+3
            float ax = Xs[buf][m * XS_LD + ka];
            float ay = Xs[buf][m * XS_LD + ka + 1];
            if (RELU) { ax = fmaxf(ax, 0.0f); ay = fmaxf(ay, 0.0f); }
            v2f a, b;
            a.x = ax; a.y = ay;
            b.x = Ws[ka * WS_LD + n_in_grp];
            b.y = Ws[(ka + 1) * WS_LD + n_in_grp];
            acc = __builtin_amdgcn_wmma_f32_16x16x4_f32(
                      false, a, false, b, (short)0, acc, false, false);
        }
        #pragma unroll
        for (int r = 0; r < 8; ++r) {
            int row = t * 16 + r + half * 8;    // C/D layout: VGPR r, half selects M+8
            H[(size_t)row * CH + cg * 64 + n_in_grp] = acc[r];
        }
        // next iteration's top barrier separates these reads from the TDM that
        // will overwrite Xs[buf] (issued at t+1)
    }
}

// -------- agg init: self-loop term + bias: agg[i,c] = b[c] + hw[i,c]*dinv[i]^2 --------
__global__ __launch_bounds__(256)
void k_agg_init(const float* __restrict__ hw, const float* __restrict__ dinv,
                const float* __restrict__ bias, float* __restrict__ agg)
{
    int idx = blockIdx.x * 256 + threadIdx.x;   // < 12.8M
    if (idx >= N_NODES * CH) return;
    int i = idx >> 7, c = idx & 127;
    float di = dinv[i];
    agg[idx] = bias[c] + hw[idx] * di * di;
}

// -------- edge scatter: one wave per edge, float4 gather + 4 atomics/lane --------
__global__ __launch_bounds__(256)
void k_edge_scatter(const float* __restrict__ hw, const int* __restrict__ src,
                    const int* __restrict__ dst, const float* __restrict__ dinv,
                    float* __restrict__ agg)
{
    int e = blockIdx.x * 8 + (threadIdx.x >> 5);
    if (e >= N_EDGES) return;
    int lane = threadIdx.x & 31;
    int s = src[e], d = dst[e];
    float w = dinv[s] * dinv[d];
    const float4* hs = (const float4*)(hw + (size_t)s * CH);
    float4 v = hs[lane];                          // 32 lanes * 16B = 512B row
    float* ap = agg + (size_t)d * CH + lane * 4;
    atomicAdd(ap + 0, v.x * w);
    atomicAdd(ap + 1, v.y * w);
    atomicAdd(ap + 2, v.z * w);
    atomicAdd(ap + 3, v.w * w);
}

// -------- output layer: 128->16 WMMA (+bias in C) fused with row softmax --------
__global__ __launch_bounds__(256)
void k_out_softmax(const float* __restrict__ Hin, const float* __restrict__ Wout,
                   const float* __restrict__ bout, float* __restrict__ out)
{
    __shared__ float Ws[128 * N_CLASSES];   // 8 KB
    const int tid  = threadIdx.x;
    const int wave = tid >> 5;
    const int lane = tid & 31;
    const int half = lane >> 4;
    const int m    = lane & 15;             // class index / row-in-tile

    for (int idx = tid; idx < 128 * N_CLASSES; idx += 256) Ws[idx] = Wout[idx];
    __syncthreads();

    int tile = blockIdx.x * 8 + wave;       // wave-uniform guard: EXEC stays all-ones
    if (tile >= M_TILES) return;

    float bval = bout[m];
    v8f acc;
    #pragma unroll
    for (int r = 0; r < 8; ++r) acc[r] = bval;

    #pragma unroll
    for (int k0 = 0; k0 < 128; k0 += 4) {
        const int ka = k0 + half * 2;
        v2f a, b;
        const float* xr = Hin + (size_t)(tile * 16 + m) * CH + ka;
        a.x = fmaxf(xr[0], 0.0f);           // fused ReLU of layer-4 output
        a.y = fmaxf(xr[1], 0.0f);
        b.x = Ws[ka * N_CLASSES + m];
        b.y = Ws[(ka + 1) * N_CLASSES + m];
        acc = __builtin_amdgcn_wmma_f32_16x16x4_f32(
                  false, a, false, b, (short)0, acc, false, false);
    }

    #pragma unroll
    for (int r = 0; r < 8; ++r) {
        int row = tile * 16 + r + half * 8;
        float v = acc[r];
        float mx = v;
        #pragma unroll
        for (int o = 8; o >= 1; o >>= 1) mx = fmaxf(mx, __shfl_xor(mx, o, 32));
        float e = __expf(v - mx);
        float s = e;
        #pragma unroll
        for (int o = 8; o >= 1; o >>= 1) s += __shfl_xor(s, o, 32);
        out[(size_t)row * N_CLASSES + m] = e / s;
    }
}

// ---------------------------------------------------------------------------
extern "C" void kernel_launch(void* const* d_in, const int* in_sizes, int n_in,
                              void* d_out, int out_size, void* d_ws, size_t ws_size,
                              hipStream_t stream)
{
    const float* x    = (const float*)d_in[0];
    const int*   eidx = (const int*)d_in[1];      // (2, E): row0=src, row1=dst
    const float* Wl[4] = {(const float*)d_in[2], (const float*)d_in[4],
                          (const float*)d_in[6], (const float*)d_in[8]};
    const float* bl[4] = {(const float*)d_in[3], (const float*)d_in[5],
                          (const float*)d_in[7], (const float*)d_in[9]};
    const float* Wout = (const float*)d_in[10];
    const float* bout = (const float*)d_in[11];

    const int* src = eidx;
    const int* dst = eidx + N_EDGES;

    float* ws   = (float*)d_ws;
    float* dinv = ws;                               // N_NODES
    float* hw   = ws + (1 << 17);                   // N_NODES*CH (gemm output)
    float* h    = hw + (size_t)N_NODES * CH;        // N_NODES*CH (aggregated)
    float* out  = (float*)d_out;

    // degree / norm
    k_deg_init<<<(N_NODES + 255) / 256, 256, 0, stream>>>(dinv);
    k_deg_scatter<<<(N_EDGES + 255) / 256, 256, 0, stream>>>(dst, dinv);
    k_deg_fin<<<(N_NODES + 255) / 256, 256, 0, stream>>>(dinv);

    // 4 GCN layers
    dim3 ggrid((M_TILES + GTPB - 1) / GTPB, 2);
    const float* hin = x;
    for (int l = 0; l < 4; ++l) {
        if (l == 0) k_gemm128<0><<<ggrid, 128, 0, stream>>>(hin, Wl[l], hw);
        else        k_gemm128<1><<<ggrid, 128, 0, stream>>>(hin, Wl[l], hw);
        k_agg_init<<<(N_NODES * CH + 255) / 256, 256, 0, stream>>>(hw, dinv, bl[l], h);
        k_edge_scatter<<<(N_EDGES + 7) / 8, 256, 0, stream>>>(hw, src, dst, dinv, h);
        hin = h;
    }

    // output projection + softmax
    k_out_softmax<<<(M_TILES + 7) / 8, 256, 0, stream>>>(h, Wout, bout, out);
}